// GCN_ancestor_edges_38981123179102
// MI455X (gfx1250) — compile-verified
//
#include <hip/hip_runtime.h>
#include <hip/hip_bf16.h>

typedef __attribute__((ext_vector_type(2))) float v2f;
typedef __attribute__((ext_vector_type(8))) float v8f;

#define KPAD 132   // max padded K (F up to 130, rounded to multiple of 4)

// ---------------------------------------------------------------------------
// h[N,16] = x[N,F] @ W[F,16] via V_WMMA_F32_16X16X4_F32, one wave per 16-row
// tile. W is staged transposed + zero-padded in LDS once per block, so the
// inner loop is 2 unguarded global b32 loads (A), one aligned ds_load_b64 (B),
// and one WMMA. Tail K (F=129/130) handled by one branchless iteration with
// clamped addresses + value selects (no exec churn).
// A f32 16x4 layout: lane L holds M=L&15, K=2*(L>>4)+{0,1}.
// B f32 4x16 layout (mirror): lane L holds N=L&15, K=2*(L>>4)+{0,1}.
// C/D: VGPR r holds M = r + 8*(L>=16), N = L&15.
// ---------------------------------------------------------------------------
__global__ void gemm_xw_wmma(const float* __restrict__ x, const float* __restrict__ W,
                             float* __restrict__ h, int n, int F) {
    __shared__ float Wt[16][KPAD];          // transposed, zero-padded
    int Fpad = (F + 3) & ~3;
    for (int i = threadIdx.x; i < 16 * KPAD; i += blockDim.x) {
        int nn = i / KPAD, k = i - nn * KPAD;
        Wt[nn][k] = (k < F) ? W[(size_t)k * 16 + nn] : 0.0f;
    }
    __syncthreads();

    int wave = (int)((blockIdx.x * blockDim.x + threadIdx.x) >> 5);
    int lane = (int)(threadIdx.x & 31);
    int row0 = wave * 16;
    if (row0 >= n) return;                   // wave-uniform exit (after barrier)

    int m  = lane & 15;
    int kk = (lane >> 4) * 2;
    const float* xrow = x + (size_t)(row0 + m) * (size_t)F;
    v8f c = {};

    int Ffull = F & ~3;                      // unguarded main loop bound
    #pragma unroll 4
    for (int k0 = 0; k0 < Ffull; k0 += 4) {
        int ka = k0 + kk;
        v2f a, b;
        a.x = xrow[ka];
        a.y = xrow[ka + 1];
        float2 tb = *(const float2*)(&Wt[m][ka]);   // 8B-aligned LDS b64
        b.x = tb.x; b.y = tb.y;
        c = __builtin_amdgcn_wmma_f32_16x16x4_f32(false, a, false, b,
                                                  (short)0, c, false, false);
    }
    if (Ffull < Fpad) {                      // uniform tail: F = 129 / 130
        int ka = Ffull + kk;
        int ka0 = (ka     < F) ? ka     : F - 1;   // clamped, always-valid loads
        int ka1 = (ka + 1 < F) ? ka + 1 : F - 1;
        float va = xrow[ka0];
        float vb = xrow[ka1];
        v2f a, b;
        a.x = (ka     < F) ? va : 0.0f;      // branchless v_cndmask
        a.y = (ka + 1 < F) ? vb : 0.0f;
        float2 tb = *(const float2*)(&Wt[m][ka]);  // LDS is zero-padded
        b.x = tb.x; b.y = tb.y;
        c = __builtin_amdgcn_wmma_f32_16x16x4_f32(false, a, false, b,
                                                  (short)0, c, false, false);
    }

    int rbase = row0 + ((lane >> 4) << 3);
    #pragma unroll
    for (int r = 0; r < 8; ++r)
        h[(size_t)(rbase + r) * 16 + m] = c[r];
}

// Final projection: h5[N,10] = relu(hf[N,16]) @ We[16,10]. B staged transposed
// + zero-padded in LDS; ReLU fused into A loads; K=16 fully unrolled.
__global__ void gemm_final_wmma(const float* __restrict__ hf, const float* __restrict__ We,
                                float* __restrict__ h5, int n) {
    __shared__ float Bs[16][16];             // Bs[n][k] = We[k][n], zero-padded
    for (int i = threadIdx.x; i < 256; i += blockDim.x) {
        int nn = i >> 4, k = i & 15;
        Bs[nn][k] = (nn < 10) ? We[(size_t)k * 10 + nn] : 0.0f;
    }
    __syncthreads();

    int wave = (int)((blockIdx.x * blockDim.x + threadIdx.x) >> 5);
    int lane = (int)(threadIdx.x & 31);
    int row0 = wave * 16;
    if (row0 >= n) return;

    int m  = lane & 15;
    int kk = (lane >> 4) * 2;
    const float* arow = hf + (size_t)(row0 + m) * 16;
    v8f c = {};
    #pragma unroll
    for (int k0 = 0; k0 < 16; k0 += 4) {
        int ka = k0 + kk;
        v2f a, b;
        a.x = fmaxf(arow[ka],     0.0f);
        a.y = fmaxf(arow[ka + 1], 0.0f);
        float2 tb = *(const float2*)(&Bs[m][ka]);
        b.x = tb.x; b.y = tb.y;
        c = __builtin_amdgcn_wmma_f32_16x16x4_f32(false, a, false, b,
                                                  (short)0, c, false, false);
    }
    int rbase = row0 + ((lane >> 4) << 3);
    if (m < 10) {
        #pragma unroll
        for (int r = 0; r < 8; ++r)
            h5[(size_t)(rbase + r) * 10 + m] = c[r];
    }
}

// ---------------------------------------------------------------------------
// Degree / normalization
// ---------------------------------------------------------------------------
__global__ void fill_ones(float* __restrict__ p, int n) {
    int i = (int)(blockIdx.x * blockDim.x + threadIdx.x);
    if (i < n) p[i] = 1.0f;
}

__global__ void deg_count(float* __restrict__ deg, const int* __restrict__ dst, int E_) {
    int e = (int)(blockIdx.x * blockDim.x + threadIdx.x);
    if (e >= E_) return;
    __builtin_prefetch(dst + e + 8192, 0, 0);   // global_prefetch_b8: stream edge list
    unsafeAtomicAdd(&deg[dst[e]], 1.0f);
}

__global__ void rsqrt_inplace(float* __restrict__ d, int n) {
    int i = (int)(blockIdx.x * blockDim.x + threadIdx.x);
    if (i < n) d[i] = rsqrtf(d[i]);
}

// ---------------------------------------------------------------------------
// Fused aggregation init: out = bias + h*dinv^2 (self loop)
//                               [+ relu(res)+res] [+ add1]
// ---------------------------------------------------------------------------
__global__ void init_agg16(float* __restrict__ out, const float* __restrict__ h,
                           const float* __restrict__ dinv, const float* __restrict__ bias,
                           const float* __restrict__ res, const float* __restrict__ add1,
                           int n16) {
    int i = (int)(blockIdx.x * blockDim.x + threadIdx.x);
    if (i >= n16) return;
    int v = i >> 4, j = i & 15;
    float d = dinv[v];
    float val = bias[j] + h[i] * d * d;
    if (res)  { float r = res[i]; val += fmaxf(r, 0.0f) + r; }
    if (add1) val += add1[i];
    out[i] = val;
}

__global__ void init_agg10(float* __restrict__ out, const float* __restrict__ h,
                           const float* __restrict__ dinv, const float* __restrict__ bias,
                           int n) {
    int i = (int)(blockIdx.x * blockDim.x + threadIdx.x);
    if (i >= n * 10) return;
    int v = i / 10, j = i - v * 10;
    float d = dinv[v];
    out[i] = bias[j] + h[i] * d * d;
}

// ---------------------------------------------------------------------------
// Edge scatter: agg[dst] += h[src] * dinv[src]*dinv[dst]
// 4 lanes per edge, float4 gather from L2-resident h, hw float atomics.
// ---------------------------------------------------------------------------
__global__ void scatter16(float* __restrict__ agg, const float* __restrict__ h,
                          const int* __restrict__ src, const int* __restrict__ dst,
                          const float* __restrict__ dinv, int E_) {
    int t = (int)(blockIdx.x * blockDim.x + threadIdx.x);
    int e = t >> 2;
    if (e >= E_) return;
    int q = (t & 3) * 4;
    int s = src[e], d = dst[e];
    float coef = dinv[s] * dinv[d];
    float4 hv = *(const float4*)(h + (size_t)s * 16 + q);
    float* ad = agg + (size_t)d * 16 + q;
    unsafeAtomicAdd(ad + 0, hv.x * coef);
    unsafeAtomicAdd(ad + 1, hv.y * coef);
    unsafeAtomicAdd(ad + 2, hv.z * coef);
    unsafeAtomicAdd(ad + 3, hv.w * coef);
}

__global__ void scatter10(float* __restrict__ agg, const float* __restrict__ h,
                          const int* __restrict__ src, const int* __restrict__ dst,
                          const float* __restrict__ dinv, int E_) {
    int t = (int)(blockIdx.x * blockDim.x + threadIdx.x);
    int e = t >> 1;
    if (e >= E_) return;
    int q = (t & 1) * 5;
    int s = src[e], d = dst[e];
    float coef = dinv[s] * dinv[d];
    const float* hs = h + (size_t)s * 10 + q;
    float* ad = agg + (size_t)d * 10 + q;
    #pragma unroll
    for (int j = 0; j < 5; ++j)
        unsafeAtomicAdd(ad + j, hs[j] * coef);
}

// In-place log-softmax over rows of 10.
__global__ void logsoftmax10(float* __restrict__ out, int n) {
    int v = (int)(blockIdx.x * blockDim.x + threadIdx.x);
    if (v >= n) return;
    float* row = out + (size_t)v * 10;
    float r[10], mx = -3.402823466e38f;
    #pragma unroll
    for (int j = 0; j < 10; ++j) { r[j] = row[j]; mx = fmaxf(mx, r[j]); }
    float s = 0.0f;
    #pragma unroll
    for (int j = 0; j < 10; ++j) s += __expf(r[j] - mx);
    float lse = mx + __logf(s);
    #pragma unroll
    for (int j = 0; j < 10; ++j) row[j] = r[j] - lse;
}

// ---------------------------------------------------------------------------
extern "C" void kernel_launch(void* const* d_in, const int* in_sizes, int n_in,
                              void* d_out, int out_size, void* d_ws, size_t ws_size,
                              hipStream_t stream) {
    const float* x_p  = (const float*)d_in[0];
    const float* x_c1 = (const float*)d_in[1];
    const float* x_c2 = (const float*)d_in[2];
    const float* x_fd = (const float*)d_in[3];
    const int* ei_p   = (const int*)d_in[4];
    const int* ei_c1  = (const int*)d_in[5];
    const int* ei_c2  = (const int*)d_in[6];
    const int* ei_fd  = (const int*)d_in[7];
    const float* W1 = (const float*)d_in[8];
    const float* b1 = (const float*)d_in[9];
    const float* W2 = (const float*)d_in[10];
    const float* b2 = (const float*)d_in[11];
    const float* W3 = (const float*)d_in[12];
    const float* b3 = (const float*)d_in[13];
    const float* We = (const float*)d_in[14];
    const float* be = (const float*)d_in[15];
    float* out = (float*)d_out;

    const int n  = out_size / 10;          // 100000
    const int E_ = in_sizes[4] / 2;        // 3.2M
    const int F1 = in_sizes[8]  / 16;      // 128
    const int F2 = in_sizes[10] / 16;      // 129
    const int F3 = in_sizes[12] / 16;      // 130

    // Workspace layout (all re-written every call)
    float* ws = (float*)d_ws;
    size_t n16 = (size_t)n * 16;
    float* hp    = ws + 0 * n16;
    float* hc1   = ws + 1 * n16;
    float* hc2   = ws + 2 * n16;
    float* hf    = ws + 3 * n16;
    float* htmp  = ws + 4 * n16;           // per-conv GEMM output (h5 for conv5)
    float* deg   = ws + 5 * n16;           // 4*n: becomes dinv after rsqrt
    float* d_p   = deg + 0 * (size_t)n;
    float* d_dc1 = deg + 1 * (size_t)n;
    float* d_dc2 = deg + 2 * (size_t)n;
    float* d_fd  = deg + 3 * (size_t)n;

    const int B = 256;
    auto cdiv = [](long long a, long long b) { return (int)((a + b - 1) / b); };
    int g_n16  = cdiv((long long)n * 16, B);
    int g_n10  = cdiv((long long)n * 10, B);
    int g_n    = cdiv(n, B);
    int g_4n   = cdiv(4LL * n, B);
    int g_E    = cdiv(E_, B);
    int g_4E   = cdiv(4LL * E_, B);
    int g_2E   = cdiv(2LL * E_, B);
    int g_gemm = cdiv((long long)((n + 15) / 16), B / 32); // one wave per tile

    // Degrees (with self loop) -> dinv, for all 4 edge lists
    fill_ones<<<g_4n, B, 0, stream>>>(deg, 4 * n);
    deg_count<<<g_E, B, 0, stream>>>(d_p,   ei_p  + E_, E_);
    deg_count<<<g_E, B, 0, stream>>>(d_dc1, ei_c1 + E_, E_);
    deg_count<<<g_E, B, 0, stream>>>(d_dc2, ei_c2 + E_, E_);
    deg_count<<<g_E, B, 0, stream>>>(d_fd,  ei_fd + E_, E_);
    rsqrt_inplace<<<g_4n, B, 0, stream>>>(deg, 4 * n);

    // conv1: hp = gcn(x_p, W1, b1)
    gemm_xw_wmma<<<g_gemm, B, 0, stream>>>(x_p, W1, htmp, n, F1);
    init_agg16<<<g_n16, B, 0, stream>>>(hp, htmp, d_p, b1, nullptr, nullptr, n * 16);
    scatter16<<<g_4E, B, 0, stream>>>(hp, htmp, ei_p, ei_p + E_, d_p, E_);

    // conv2: hc1 = gcn(x_c1, W2, b2) + relu(hp) + hp
    gemm_xw_wmma<<<g_gemm, B, 0, stream>>>(x_c1, W2, htmp, n, F2);
    init_agg16<<<g_n16, B, 0, stream>>>(hc1, htmp, d_dc1, b2, hp, nullptr, n * 16);
    scatter16<<<g_4E, B, 0, stream>>>(hc1, htmp, ei_c1, ei_c1 + E_, d_dc1, E_);

    // conv3: hc2 = gcn(x_c2, W3, b3) + relu(hc1) + hc1
    gemm_xw_wmma<<<g_gemm, B, 0, stream>>>(x_c2, W3, htmp, n, F3);
    init_agg16<<<g_n16, B, 0, stream>>>(hc2, htmp, d_dc2, b3, hc1, nullptr, n * 16);
    scatter16<<<g_4E, B, 0, stream>>>(hc2, htmp, ei_c2, ei_c2 + E_, d_dc2, E_);

    // conv4: hf = gcn(x_fd, W2, b2) + relu(hc2) + hc2 + hc1
    gemm_xw_wmma<<<g_gemm, B, 0, stream>>>(x_fd, W2, htmp, n, F2);
    init_agg16<<<g_n16, B, 0, stream>>>(hf, htmp, d_fd, b2, hc2, hc1, n * 16);
    scatter16<<<g_4E, B, 0, stream>>>(hf, htmp, ei_fd, ei_fd + E_, d_fd, E_);

    // conv5: out = log_softmax(gcn(relu(hf), We, be))
    gemm_final_wmma<<<g_gemm, B, 0, stream>>>(hf, We, htmp, n);
    init_agg10<<<g_n10, B, 0, stream>>>(out, htmp, d_fd, be, n);
    scatter10<<<g_2E, B, 0, stream>>>(out, htmp, ei_fd, ei_fd + E_, d_fd, E_);
    logsoftmax10<<<g_n, B, 0, stream>>>(out, n);
}